// LightGCN_12421045420269
// MI455X (gfx1250) — compile-verified
//
#include <hip/hip_runtime.h>

// ---------------------------------------------------------------------------
// LightGCN forward on MI455X (gfx1250, wave32)
//   3 x COO SpMM (gather + f32 atomic scatter) + fused layer-mean epilogue.
//   Epilogue runs on the WMMA f32 pipe (V_WMMA_F32_16X16X4_F32) with a
//   0.25-scaled one-hot A operand: D = (0.25*I) x Acc  (exact in f32).
// ---------------------------------------------------------------------------

#define NUM_USERS 100000
#define NUM_ITEMS 200000
#define NN        (NUM_USERS + NUM_ITEMS)   // 300000 rows
#define NEDGE     2000000
#define DIM       64

typedef float  f4  __attribute__((ext_vector_type(4)));
typedef float  v2f __attribute__((ext_vector_type(2)));
typedef float  v8f __attribute__((ext_vector_type(8)));

#define TOTAL4 (NN * DIM / 4)   // 4.8M float4 elements
#define USER4  (NUM_USERS * DIM / 4)

// ---- acc = concat(user_emb, item_emb) --------------------------------------
__global__ void lgcn_init_acc(const float* __restrict__ ue,
                              const float* __restrict__ ie,
                              float* __restrict__ acc) {
    int stride = gridDim.x * blockDim.x;
    for (int i = blockIdx.x * blockDim.x + threadIdx.x; i < TOTAL4; i += stride) {
        f4 v = (i < USER4) ? ((const f4*)ue)[i] : ((const f4*)ie)[i - USER4];
        ((f4*)acc)[i] = v;
    }
}

// ---- y = 0 -----------------------------------------------------------------
__global__ void lgcn_zero(float* __restrict__ y) {
    int stride = gridDim.x * blockDim.x;
    f4 z = {0.f, 0.f, 0.f, 0.f};
    for (int i = blockIdx.x * blockDim.x + threadIdx.x; i < TOTAL4; i += stride)
        ((f4*)y)[i] = z;
}

// ---- COO SpMM: y[r] += v * x[c]   (2 edges per wave, 4 floats per lane) ----
// layer0 gathers from the concat(user,item) inputs directly; later layers
// gather from the dense h buffer. Edge lists are streamed with NT hints so
// they never evict the L2-resident embedding table.
__global__ void lgcn_spmm(const int*   __restrict__ rows,
                          const int*   __restrict__ cols,
                          const float* __restrict__ vals,
                          const float* __restrict__ x,      // dense src (layers>0)
                          const float* __restrict__ ue,     // concat src (layer 0)
                          const float* __restrict__ ie,
                          int use_concat,
                          float* __restrict__ y) {
    const int lane = threadIdx.x & 31;
    const int sub  = lane & 15;          // float4 chunk 0..15 of the 64-f row
    const int half = lane >> 4;          // which of the wave's 2 edges
    const int wid  = (blockIdx.x * blockDim.x + threadIdx.x) >> 5;
    const int nw   = (gridDim.x * blockDim.x) >> 5;

    for (int e2 = wid; e2 * 2 < NEDGE; e2 += nw) {
        const int e = e2 * 2 + half;     // NEDGE is even -> always in range

        // prefetch the edge stream a few sweeps ahead (global_prefetch_b8)
        if (lane == 0) {
            int pe = (e2 + nw * 4) * 2;
            if (pe < NEDGE) {
                __builtin_prefetch(&rows[pe], 0, 0);
                __builtin_prefetch(&cols[pe], 0, 0);
                __builtin_prefetch(&vals[pe], 0, 0);
            }
        }

        const int   r = __builtin_nontemporal_load(&rows[e]);
        const int   c = __builtin_nontemporal_load(&cols[e]);
        const float v = __builtin_nontemporal_load(&vals[e]);

        const float* src;
        size_t off;
        if (use_concat) {
            if (c < NUM_USERS) { src = ue; off = (size_t)c * DIM; }
            else               { src = ie; off = (size_t)(c - NUM_USERS) * DIM; }
        } else {
            src = x; off = (size_t)c * DIM;
        }

        // 16 lanes x float4 = one contiguous 256B row gather per edge
        f4 d = ((const f4*)(src + off))[sub];

        float* dst = y + (size_t)r * DIM + sub * 4;
        // non-returning f32 atomics -> global_atomic_add_f32 (STOREcnt path)
        __hip_atomic_fetch_add(dst + 0, v * d.x, __ATOMIC_RELAXED, __HIP_MEMORY_SCOPE_AGENT);
        __hip_atomic_fetch_add(dst + 1, v * d.y, __ATOMIC_RELAXED, __HIP_MEMORY_SCOPE_AGENT);
        __hip_atomic_fetch_add(dst + 2, v * d.z, __ATOMIC_RELAXED, __HIP_MEMORY_SCOPE_AGENT);
        __hip_atomic_fetch_add(dst + 3, v * d.w, __ATOMIC_RELAXED, __HIP_MEMORY_SCOPE_AGENT);
    }
}

// ---- acc += h  (h is read exactly once -> NT loads) ------------------------
__global__ void lgcn_acc_add(float* __restrict__ acc, const float* __restrict__ h) {
    int stride = gridDim.x * blockDim.x;
    const f4* hp = (const f4*)h;
    for (int i = blockIdx.x * blockDim.x + threadIdx.x; i < TOTAL4; i += stride) {
        f4 a = ((f4*)acc)[i];
        f4 b = __builtin_nontemporal_load(&hp[i]);
        a.x += b.x; a.y += b.y; a.z += b.z; a.w += b.w;
        ((f4*)acc)[i] = a;
    }
}

// ---- epilogue: acc *= 0.25 via V_WMMA_F32_16X16X4_F32 ----------------------
// Per 16x16 tile:  D = sum_{j=0..3} A_j x B_j,  A_j[m,k] = 0.25 * (m == 4j+k),
// B_j = rows 4j..4j+3 of the tile. Exact (0.25 and f32 adds), in-place, and
// runs on the matrix pipe so the VALU stays free on this BW-bound tail.
// Operand layouts per ISA §7.12.2:
//   A 16x4 : lane L holds M=L%16; VGPR{0,1} = K{0,1} (L<16) / K{2,3} (L>=16)
//   B 4x16 : lane L holds N=L%16; VGPR{0,1} = K{0,1} (L<16) / K{2,3} (L>=16)
//   C/D    : lane L holds N=L%16; VGPR i    = M = i + (L<16 ? 0 : 8)
__global__ void lgcn_finalize_wmma(float* __restrict__ acc) {
    const int lane   = threadIdx.x & 31;
    const int wid    = (blockIdx.x * blockDim.x + threadIdx.x) >> 5;
    const int nw     = (gridDim.x * blockDim.x) >> 5;
    const int nTiles = (NN / 16) * (DIM / 16);   // 18750 * 4 = 75000

    const int n  = lane & 15;
    const int m  = lane & 15;
    const int kA = (lane < 16) ? 0 : 2;
    const int mB = (lane < 16) ? 0 : 8;

    for (int tile = wid; tile < nTiles; tile += nw) {   // wave-uniform loop
        const int row0 = (tile >> 2) * 16;
        const int col0 = (tile & 3) * 16;

        v8f c = {};
        #pragma unroll
        for (int j = 0; j < 4; ++j) {
            v2f a, b;
            a.x = (m == 4 * j + kA)     ? 0.25f : 0.0f;
            a.y = (m == 4 * j + kA + 1) ? 0.25f : 0.0f;
            const float* base = acc + (size_t)(row0 + 4 * j + kA) * DIM + col0 + n;
            b.x = base[0];
            b.y = base[DIM];
            c = __builtin_amdgcn_wmma_f32_16x16x4_f32(
                    /*neg_a=*/false, a, /*neg_b=*/false, b,
                    /*c_mod=*/(short)0, c, /*reuse_a=*/false, /*reuse_b=*/false);
        }

        #pragma unroll
        for (int i = 0; i < 8; ++i)
            acc[(size_t)(row0 + mB + i) * DIM + col0 + n] = c[i];
    }
}

// ---------------------------------------------------------------------------
extern "C" void kernel_launch(void* const* d_in, const int* in_sizes, int n_in,
                              void* d_out, int out_size, void* d_ws, size_t ws_size,
                              hipStream_t stream) {
    const float* ue   = (const float*)d_in[0];
    const float* ie   = (const float*)d_in[1];
    const int*   rows = (const int*)  d_in[2];
    const int*   cols = (const int*)  d_in[3];
    const float* vals = (const float*)d_in[4];
    float* out = (float*)d_out;                 // [NN, 64] = users ++ items

    // workspace: two ping-pong h buffers, 76.8 MB each
    float* hA = (float*)d_ws;
    float* hB = hA + (size_t)NN * DIM;

    const int CPY_BLK = (TOTAL4 + 255) / 256;   // 1 elem/thread + stride guard
    const int SPM_BLK = 6144;                   // grid-stride over 1M edge pairs

    lgcn_init_acc<<<CPY_BLK, 256, 0, stream>>>(ue, ie, out);

    const float* x = nullptr;
    float* y = hA;
    for (int l = 0; l < 3; ++l) {
        lgcn_zero<<<CPY_BLK, 256, 0, stream>>>(y);
        lgcn_spmm<<<SPM_BLK, 256, 0, stream>>>(rows, cols, vals, x, ue, ie,
                                               l == 0 ? 1 : 0, y);
        lgcn_acc_add<<<CPY_BLK, 256, 0, stream>>>(out, y);
        x = y;
        y = (y == hA) ? hB : hA;
    }

    // 75000 tiles, 4 waves/block -> exact cover, EXEC all-ones per wave
    lgcn_finalize_wmma<<<75000 / 4, 128, 0, stream>>>(out);
}